// E3Convolution_68642167324710
// MI455X (gfx1250) — compile-verified
//
#include <hip/hip_runtime.h>
#include <math.h>

typedef float v2f __attribute__((ext_vector_type(2)));
typedef float v8f __attribute__((ext_vector_type(8)));

#define N_NODES 5000
#define N_EDGES 30000
#define WN      6912     // Wm2 columns
#define NCHUNK  216      // WN / 32  (two 16-col WMMA tiles per chunk)

// constants
#define RS32   0.17677669529663687f   // 1/sqrt(32)
#define RS16   0.25f                  // 1/sqrt(16)
#define RS128  0.08838834764831845f   // 1/sqrt(128)
#define RS64   0.125f                 // 1/sqrt(64)
#define RS1280 0.027950849718747374f  // 1/sqrt(1280)
#define RS640  0.03952847075210474f   // 1/sqrt(640)
#define RS3    0.57735026918962576f   // 1/sqrt(3)
#define FSCALE 0.007365695637359871f  // (1/sqrt(128)) * (1/sqrt(144))

#if __has_builtin(__builtin_amdgcn_s_wait_asynccnt)
#define WAIT_ASYNC_8() __builtin_amdgcn_s_wait_asynccnt(8)
#define WAIT_ASYNC_0() __builtin_amdgcn_s_wait_asynccnt(0)
#else
#define WAIT_ASYNC_8() asm volatile("s_wait_asynccnt 0x8" ::: "memory")
#define WAIT_ASYNC_0() asm volatile("s_wait_asynccnt 0x0" ::: "memory")
#endif

__device__ __forceinline__ float sigm(float x) { return 1.f / (1.f + __expf(-x)); }

// async DMA one 16-byte segment from global into LDS (ASYNCcnt-tracked)
__device__ __forceinline__ void async_b128(unsigned lds_off, const void* gaddr) {
  asm volatile("global_load_async_to_lds_b128 %0, %1, off"
               :: "v"(lds_off), "v"(gaddr)
               : "memory");
}

// ---------------------------------------------------------------------------
// Kernel A: per-node sc_tp + o3_linear(L1n), plus zero node accumulator
// ---------------------------------------------------------------------------
__global__ void k_node_pre(const float* __restrict__ f_node,
                           const float* __restrict__ node_emb,
                           const float* __restrict__ W0,   // (32,4,32)
                           const float* __restrict__ W1,   // (16,4,16)
                           const float* __restrict__ L0,   // (32,32)
                           const float* __restrict__ L1,   // (16,16)
                           float* __restrict__ fn_lin1,
                           float* __restrict__ sc_node,
                           float* __restrict__ nacc)
{
  int g = blockIdx.x * blockDim.x + threadIdx.x;
  int stride = gridDim.x * blockDim.x;
  for (int i = g; i < N_NODES * 80; i += stride) nacc[i] = 0.f;
  if (g >= N_NODES) return;
  const float* x = f_node + (size_t)g * 80;
  float s0 = node_emb[g*4+0], s1 = node_emb[g*4+1], s2 = node_emb[g*4+2], s3 = node_emb[g*4+3];
  float* fo = fn_lin1 + (size_t)g * 80;
  float* so = sc_node + (size_t)g * 80;
  for (int w = 0; w < 32; ++w) {
    float sc = 0.f, li = 0.f;
    for (int u = 0; u < 32; ++u) {
      const float* Wp = W0 + (u*4)*32 + w;
      float vw = s0*Wp[0] + s1*Wp[32] + s2*Wp[64] + s3*Wp[96];
      float xu = x[u];
      sc += xu * vw;
      li += xu * L0[u*32 + w];
    }
    so[w] = sc * RS128;
    fo[w] = li * RS32;
  }
  for (int w = 0; w < 16; ++w) {
    float sa0=0.f,sa1=0.f,sa2=0.f,la0=0.f,la1=0.f,la2=0.f;
    for (int u = 0; u < 16; ++u) {
      const float* Wp = W1 + (u*4)*16 + w;
      float vw = s0*Wp[0] + s1*Wp[16] + s2*Wp[32] + s3*Wp[48];
      float lw = L1[u*16 + w];
      float x0 = x[32+u*3+0], x1 = x[32+u*3+1], x2 = x[32+u*3+2];
      sa0 += x0*vw; sa1 += x1*vw; sa2 += x2*vw;
      la0 += x0*lw; la1 += x1*lw; la2 += x2*lw;
    }
    so[32+w*3+0] = sa0*RS64; so[32+w*3+1] = sa1*RS64; so[32+w*3+2] = sa2*RS64;
    fo[32+w*3+0] = la0*RS16; fo[32+w*3+1] = la1*RS16; fo[32+w*3+2] = la2*RS16;
  }
}

// ---------------------------------------------------------------------------
// Kernel B: per-edge sc_tp (W_sce) + o3_linear(L1e).  One wave32 per edge.
// ---------------------------------------------------------------------------
__global__ void k_edge_pre(const float* __restrict__ f_edge,
                           const float* __restrict__ node_emb,
                           const float* __restrict__ lemb,
                           const int*   __restrict__ esrc,
                           const int*   __restrict__ edst,
                           const float* __restrict__ W0,   // (32,40,32)
                           const float* __restrict__ W1,   // (16,40,16)
                           const float* __restrict__ L0,   // (32,32)
                           const float* __restrict__ L1,   // (16,16)
                           float* __restrict__ fe_lin1,
                           float* __restrict__ sc_edge)
{
  const int lane = threadIdx.x & 31;
  const int e = blockIdx.x * 8 + (threadIdx.x >> 5);
  const float* xe = f_edge + (size_t)e * 80;
  float s_[40];
  {
    int s = esrc[e], d = edst[e];
    #pragma unroll
    for (int t = 0; t < 4; ++t) s_[t] = node_emb[s*4 + t];
    #pragma unroll
    for (int t = 0; t < 4; ++t) s_[4+t] = node_emb[d*4 + t];
    #pragma unroll
    for (int t = 0; t < 32; ++t) s_[8+t] = lemb[(size_t)e*32 + t];
  }
  // scalar irreps: lane == output w
  {
    float sc = 0.f, li = 0.f;
    for (int u = 0; u < 32; ++u) {
      const float* wr = W0 + (size_t)u*40*32 + lane;
      float vw = 0.f;
      #pragma unroll 8
      for (int t = 0; t < 40; ++t) vw += s_[t] * wr[t*32];
      float xu = xe[u];
      sc += xu * vw;
      li += xu * L0[u*32 + lane];
    }
    sc_edge[(size_t)e*80 + lane] = sc * RS1280;
    fe_lin1[(size_t)e*80 + lane] = li * RS32;
  }
  // vector irreps: lanes 0..15, lane == output w, all three k components
  if (lane < 16) {
    float a0=0.f,a1=0.f,a2=0.f,l0=0.f,l1=0.f,l2=0.f;
    for (int u = 0; u < 16; ++u) {
      const float* wr = W1 + (size_t)u*40*16 + lane;
      float vw = 0.f;
      #pragma unroll 8
      for (int t = 0; t < 40; ++t) vw += s_[t] * wr[t*16];
      float x0 = xe[32+u*3+0], x1 = xe[32+u*3+1], x2 = xe[32+u*3+2];
      a0 += x0*vw; a1 += x1*vw; a2 += x2*vw;
      float lw = L1[u*16 + lane];
      l0 += x0*lw; l1 += x1*lw; l2 += x2*lw;
    }
    size_t o = (size_t)e*80 + 32 + lane*3;
    sc_edge[o+0] = a0*RS640; sc_edge[o+1] = a1*RS640; sc_edge[o+2] = a2*RS640;
    fe_lin1[o+0] = l0*RS16;  fe_lin1[o+1] = l1*RS16;  fe_lin1[o+2] = l2*RS16;
  }
}

// ---------------------------------------------------------------------------
// Kernel H: h = silu(length_emb @ Wm1 / sqrt(32))  via WMMA f32 16x16x4
// ---------------------------------------------------------------------------
__global__ void k_hidden(const float* __restrict__ lemb,
                         const float* __restrict__ Wm1,   // (32,128)
                         float* __restrict__ h)
{
  __shared__ float w1[32 * 128];
  const int tid = threadIdx.x;
  const int lane = tid & 31;
  const int wv = tid >> 5;
  const int jc = lane & 15;
  const int koff = (lane < 16) ? 0 : 2;
  const int lo = (lane < 16) ? 0 : 8;
  for (int i = tid; i < 4096; i += 128) w1[i] = Wm1[i];
  int eA = blockIdx.x * 64 + wv * 16 + jc;
  if (eA >= N_EDGES) eA = N_EDGES - 1;
  const float* le = lemb + (size_t)eA * 32;
  v2f A[8];
  #pragma unroll
  for (int kk = 0; kk < 8; ++kk) { A[kk].x = le[kk*4 + koff]; A[kk].y = le[kk*4 + koff + 1]; }
  __syncthreads();
  const int ebase = blockIdx.x * 64 + wv * 16;
  for (int nt = 0; nt < 8; ++nt) {
    v8f c = {0.f,0.f,0.f,0.f,0.f,0.f,0.f,0.f};
    #pragma unroll
    for (int kk = 0; kk < 8; ++kk) {
      v2f b;
      b.x = w1[(kk*4 + koff)     * 128 + nt*16 + jc];
      b.y = w1[(kk*4 + koff + 1) * 128 + nt*16 + jc];
      c = __builtin_amdgcn_wmma_f32_16x16x4_f32(false, A[kk], false, b, (short)0, c, false, false);
    }
    #pragma unroll
    for (int v = 0; v < 8; ++v) {
      int e = ebase + v + lo;
      if (e < N_EDGES) {
        float x = c[v] * RS32;
        h[(size_t)e * 128 + nt*16 + jc] = x * sigm(x);
      }
    }
  }
}

// ---------------------------------------------------------------------------
// Kernel C: fused  w = h @ Wm2  (WMMA, never materialized) -> e3tp -> gate
//           -> o3_linear(L2e) + sc_edge -> fe out; gated feats scattered to nacc
// block = 128 threads (4 waves), 64 edges per block, 16 edges per wave.
// Wm2 streamed in 32-column chunks via async global->LDS DMA, double-buffered.
// ---------------------------------------------------------------------------

// stage chunk m (cols [32m, 32m+32), all 128 K rows) into LDS buffer.
// buffer layout: [0,2048) = even tile rows of 16 floats, [2048,4096) = odd tile.
__device__ __forceinline__ void stage_chunk(const float* __restrict__ Wm2,
                                            int m, float* buf, int tid) {
  const int i    = tid & 7;        // 8 x 16B segments per 32-float row
  const int k0   = tid >> 3;       // rows 0..15 (+16j)
  const int tsel = i >> 2;         // which 16-col tile
  const int cof  = (i & 3) * 4;    // float offset within tile row
  const float* gbase = Wm2 + (size_t)m * 32 + tsel * 16 + cof;
  unsigned lbase = (unsigned)(size_t)(buf + tsel * 2048 + cof);
  #pragma unroll
  for (int j = 0; j < 8; ++j) {
    int k = k0 + 16 * j;
    async_b128(lbase + (unsigned)(k * 16) * 4u, gbase + (size_t)k * WN);
  }
}

__global__ void k_edge_main(const float* __restrict__ sh,
                            const int*   __restrict__ esrc,
                            const int*   __restrict__ edst,
                            const float* __restrict__ fn1,
                            const float* __restrict__ fe1,
                            const float* __restrict__ sce,
                            const float* __restrict__ h,
                            const float* __restrict__ Wm2,  // (128, 6912)
                            const float* __restrict__ L2e0, // (32,32)
                            const float* __restrict__ L2e1, // (16,16)
                            float* __restrict__ nacc,
                            float* __restrict__ feo)
{
  extern __shared__ float sm[];
  float* smB   = sm;               // 8192 : double-buffered 128x32 Wm2 chunk
  float* smAss = sm + 8192;        // 6144 : [96][64]  x0cat * sh0
  float* smAsv = sm + 14336;       // 6144 : [96][64]  x0cat
  float* smAvv = sm + 20480;       // 3072 : [48][64]  (x1cat . sh1)/sqrt3
  float* smAvs = sm + 23552;       // 9216 : [3][48][64] x1cat components
  float* smL0  = sm + 32768;       // 1024
  float* smL1  = sm + 33792;       // 256   (total 34048 floats)
  float* smStage = sm;             // 6144 : reused after main loop

  const int tid  = threadIdx.x;
  const int lane = tid & 31;
  const int wv   = tid >> 5;
  const int jc   = lane & 15;
  const int koff = (lane < 16) ? 0 : 2;
  const int lo   = (lane < 16) ? 0 : 8;
  const int eblk = blockIdx.x * 64;
  const int ew   = wv * 16;

  for (int i = tid; i < 1024; i += 128) smL0[i] = L2e0[i];
  for (int i = tid; i < 256;  i += 128) smL1[i] = L2e1[i];

  // -------- per-edge gathered features -> LDS (threads 0..63, one edge each)
  if (tid < 64) {
    int e = eblk + tid; if (e >= N_EDGES) e = N_EDGES - 1;
    const int s = esrc[e], d = edst[e];
    const float sh0 = sh[e*4+0], s1x = sh[e*4+1], s1y = sh[e*4+2], s1z = sh[e*4+3];
    const float* fs = fn1 + (size_t)s * 80;
    const float* fd = fn1 + (size_t)d * 80;
    const float* fg = fe1 + (size_t)e * 80;
    #pragma unroll 4
    for (int u = 0; u < 32; ++u) {
      float a = fs[u], b = fd[u], c = fg[u];
      smAsv[u*64 + tid]      = a;     smAss[u*64 + tid]      = a * sh0;
      smAsv[(u+32)*64 + tid] = b;     smAss[(u+32)*64 + tid] = b * sh0;
      smAsv[(u+64)*64 + tid] = c;     smAss[(u+64)*64 + tid] = c * sh0;
    }
    #pragma unroll 2
    for (int u = 0; u < 16; ++u) {
      float ax = fs[32+u*3+0], ay = fs[32+u*3+1], az = fs[32+u*3+2];
      float bx = fd[32+u*3+0], by = fd[32+u*3+1], bz = fd[32+u*3+2];
      float cx = fg[32+u*3+0], cy = fg[32+u*3+1], cz = fg[32+u*3+2];
      smAvs[u*64 + tid]        = ax; smAvs[3072 + u*64 + tid]        = ay; smAvs[6144 + u*64 + tid]        = az;
      smAvs[(u+16)*64 + tid]   = bx; smAvs[3072 + (u+16)*64 + tid]   = by; smAvs[6144 + (u+16)*64 + tid]   = bz;
      smAvs[(u+32)*64 + tid]   = cx; smAvs[3072 + (u+32)*64 + tid]   = cy; smAvs[6144 + (u+32)*64 + tid]   = cz;
      smAvv[u*64 + tid]      = (ax*s1x + ay*s1y + az*s1z) * RS3;
      smAvv[(u+16)*64 + tid] = (bx*s1x + by*s1y + bz*s1z) * RS3;
      smAvv[(u+32)*64 + tid] = (cx*s1x + cy*s1y + cz*s1z) * RS3;
    }
  }

  // -------- A fragments: h rows for this wave's 16 edges, all K=128 (64 VGPRs)
  int eA = eblk + ew + jc; if (eA >= N_EDGES) eA = N_EDGES - 1;
  const float* hr = h + (size_t)eA * 128;
  v2f A[32];
  #pragma unroll
  for (int kk = 0; kk < 32; ++kk) { A[kk].x = hr[kk*4 + koff]; A[kk].y = hr[kk*4 + koff + 1]; }

  v8f Y0a = {0.f,0.f,0.f,0.f,0.f,0.f,0.f,0.f};
  v8f Y0b = Y0a, T1 = Y0a, T2x = Y0a, T2y = Y0a, T2z = Y0a;

  // prime the async pipeline with chunk 0
  stage_chunk(Wm2, 0, smB, tid);
  __syncthreads();   // also covers the feature-LDS writes above

  // -------- main loop over 216 chunks (2 WMMA N-tiles each)
  for (int m = 0; m < NCHUNK; ++m) {
    float* bcur = smB + (m & 1) * 4096;
    if (m + 1 < NCHUNK) {
      stage_chunk(Wm2, m + 1, smB + ((m + 1) & 1) * 4096, tid);
      WAIT_ASYNC_8();            // newest 8 (chunk m+1) may remain in flight
    } else {
      WAIT_ASYNC_0();
    }
    __syncthreads();             // chunk m fully resident for all waves

    v8f c0 = {0.f,0.f,0.f,0.f,0.f,0.f,0.f,0.f};
    v8f c1 = {0.f,0.f,0.f,0.f,0.f,0.f,0.f,0.f};
    #pragma unroll
    for (int kk = 0; kk < 32; ++kk) {
      const float* br0 = bcur + (kk*4 + koff) * 16;
      const float* br1 = br0 + 2048;
      v2f b0; b0.x = br0[jc]; b0.y = br0[16 + jc];
      v2f b1; b1.x = br1[jc]; b1.y = br1[16 + jc];
      c0 = __builtin_amdgcn_wmma_f32_16x16x4_f32(false, A[kk], false, b0, (short)0, c0, false, false);
      c1 = __builtin_amdgcn_wmma_f32_16x16x4_f32(false, A[kk], false, b1, (short)0, c1, false, false);
    }

    // contraction: chunk m = tiles 2m, 2m+1
    const int eix = ew + lo;
    if (m < 96) {                        // w_ss -> y0 : u = m, halves 0/1
      const float* a = smAss + m*64 + eix;
      #pragma unroll
      for (int v = 0; v < 8; ++v) { float av = a[v]; Y0a[v] += av * c0[v]; Y0b[v] += av * c1[v]; }
    } else if (m < 144) {                // w_sv -> t1 : u = 2m-192, 2m-191
      const float* a0 = smAsv + (2*m - 192)*64 + eix;
      const float* a1 = a0 + 64;
      #pragma unroll
      for (int v = 0; v < 8; ++v) T1[v] += a0[v]*c0[v] + a1[v]*c1[v];
    } else if (m < 168) {                // w_vs -> t2[k] : u = 2m-288, 2m-287
      int u0 = 2*m - 288;
      const float* ax0 = smAvs + u0*64 + eix;          const float* ax1 = ax0 + 64;
      const float* ay0 = smAvs + 3072 + u0*64 + eix;   const float* ay1 = ay0 + 64;
      const float* az0 = smAvs + 6144 + u0*64 + eix;   const float* az1 = az0 + 64;
      #pragma unroll
      for (int v = 0; v < 8; ++v) {
        T2x[v] += ax0[v]*c0[v] + ax1[v]*c1[v];
        T2y[v] += ay0[v]*c0[v] + ay1[v]*c1[v];
        T2z[v] += az0[v]*c0[v] + az1[v]*c1[v];
      }
    } else {                             // w_vv -> y0 : u = m-168, halves 0/1
      const float* a = smAvv + (m - 168)*64 + eix;
      #pragma unroll
      for (int v = 0; v < 8; ++v) { float av = a[v]; Y0a[v] += av * c0[v]; Y0b[v] += av * c1[v]; }
    }
    __syncthreads();             // protect buffer reuse by next iteration's DMA
  }

  // -------- stage accumulators to LDS (per-edge layout), then epilogue
  #pragma unroll
  for (int v = 0; v < 8; ++v) {
    float* st = smStage + (ew + lo + v) * 96;
    st[jc]      = Y0a[v];
    st[16 + jc] = Y0b[v];
    st[32 + jc] = T1[v];
    st[48 + jc] = T2x[v];
    st[64 + jc] = T2y[v];
    st[80 + jc] = T2z[v];
  }
  __syncthreads();

  if (tid < 64) {
    int e = eblk + tid;
    if (e < N_EDGES) {
      const float* stg = smStage + tid * 96;
      const float sh0 = sh[e*4+0], s1x = sh[e*4+1], s1y = sh[e*4+2], s1z = sh[e*4+3];
      float g0[32];
      #pragma unroll
      for (int w = 0; w < 32; ++w) {
        float y = stg[w] * FSCALE;
        g0[w] = y * sigm(y);               // silu
      }
      float g1[16][3];
      #pragma unroll
      for (int w = 0; w < 16; ++w) {
        float t1v = stg[32 + w];
        float yx = (t1v*s1x + sh0*stg[48 + w]) * FSCALE;
        float yy = (t1v*s1y + sh0*stg[64 + w]) * FSCALE;
        float yz = (t1v*s1z + sh0*stg[80 + w]) * FSCALE;
        float sg = sigm(sqrtf(yx*yx + yy*yy + yz*yz));
        g1[w][0] = yx*sg; g1[w][1] = yy*sg; g1[w][2] = yz*sg;
      }
      // scatter gated features to destination node accumulator
      float* np = nacc + (size_t)edst[e] * 80;
      #pragma unroll
      for (int w = 0; w < 32; ++w) atomicAdd(np + w, g0[w]);
      #pragma unroll
      for (int w = 0; w < 16; ++w) {
        atomicAdd(np + 32 + w*3 + 0, g1[w][0]);
        atomicAdd(np + 32 + w*3 + 1, g1[w][1]);
        atomicAdd(np + 32 + w*3 + 2, g1[w][2]);
      }
      // o3_linear(L2e) + sc_edge -> fe output
      const float* sc = sce + (size_t)e * 80;
      float* out = feo + (size_t)e * 80;
      for (int w = 0; w < 32; ++w) {
        float acc = 0.f;
        #pragma unroll 8
        for (int u = 0; u < 32; ++u) acc += g0[u] * smL0[u*32 + w];
        out[w] = acc * RS32 + sc[w];
      }
      for (int w = 0; w < 16; ++w) {
        float ax = 0.f, ay = 0.f, az = 0.f;
        #pragma unroll 8
        for (int u = 0; u < 16; ++u) {
          float l = smL1[u*16 + w];
          ax += g1[u][0]*l; ay += g1[u][1]*l; az += g1[u][2]*l;
        }
        out[32 + w*3 + 0] = ax * RS16 + sc[32 + w*3 + 0];
        out[32 + w*3 + 1] = ay * RS16 + sc[32 + w*3 + 1];
        out[32 + w*3 + 2] = az * RS16 + sc[32 + w*3 + 2];
      }
    }
  }
}

// ---------------------------------------------------------------------------
// Kernel D: fn = o3_linear(nacc/6, L2n) + sc_node
// ---------------------------------------------------------------------------
__global__ void k_node_post(const float* __restrict__ nacc,
                            const float* __restrict__ sc_node,
                            const float* __restrict__ L0,
                            const float* __restrict__ L1,
                            float* __restrict__ out)
{
  int g = blockIdx.x * blockDim.x + threadIdx.x;
  if (g >= N_NODES) return;
  const float* a  = nacc + (size_t)g * 80;
  const float* sc = sc_node + (size_t)g * 80;
  float* o = out + (size_t)g * 80;
  const float inv6 = 1.f / 6.f;
  for (int w = 0; w < 32; ++w) {
    float acc = 0.f;
    for (int u = 0; u < 32; ++u) acc += a[u] * L0[u*32 + w];
    o[w] = acc * inv6 * RS32 + sc[w];
  }
  for (int w = 0; w < 16; ++w) {
    float ax = 0.f, ay = 0.f, az = 0.f;
    for (int u = 0; u < 16; ++u) {
      float l = L1[u*16 + w];
      ax += a[32+u*3+0]*l; ay += a[32+u*3+1]*l; az += a[32+u*3+2]*l;
    }
    o[32+w*3+0] = ax*inv6*RS16 + sc[32+w*3+0];
    o[32+w*3+1] = ay*inv6*RS16 + sc[32+w*3+1];
    o[32+w*3+2] = az*inv6*RS16 + sc[32+w*3+2];
  }
}

// ---------------------------------------------------------------------------
extern "C" void kernel_launch(void* const* d_in, const int* in_sizes, int n_in,
                              void* d_out, int out_size, void* d_ws, size_t ws_size,
                              hipStream_t stream) {
  const float* f_node     = (const float*)d_in[0];
  const float* f_edge     = (const float*)d_in[1];
  const float* sh         = (const float*)d_in[2];
  const float* node_emb   = (const float*)d_in[3];
  const float* length_emb = (const float*)d_in[4];
  const float* W_scn0     = (const float*)d_in[5];
  const float* W_scn1     = (const float*)d_in[6];
  const float* W_sce0     = (const float*)d_in[7];
  const float* W_sce1     = (const float*)d_in[8];
  const float* L1n0       = (const float*)d_in[9];
  const float* L1n1       = (const float*)d_in[10];
  const float* L1e0       = (const float*)d_in[11];
  const float* L1e1       = (const float*)d_in[12];
  const float* Wm1        = (const float*)d_in[13];
  const float* Wm2        = (const float*)d_in[14];
  const float* L2n0       = (const float*)d_in[15];
  const float* L2n1       = (const float*)d_in[16];
  const float* L2e0       = (const float*)d_in[17];
  const float* L2e1       = (const float*)d_in[18];
  const int*   esrc       = (const int*)d_in[19];
  const int*   edst       = (const int*)d_in[20];
  float* out = (float*)d_out;
  float* ws  = (float*)d_ws;

  float* fn_lin1 = ws;                 //   400000
  float* sc_node = ws + 400000;        //   400000
  float* nacc    = ws + 800000;        //   400000
  float* fe_lin1 = ws + 1200000;       //  2400000
  float* sc_edge = ws + 3600000;       //  2400000
  float* hbuf    = ws + 6000000;       //  3840000   (total 9.84M floats)

  k_node_pre<<<40, 128, 0, stream>>>(f_node, node_emb, W_scn0, W_scn1, L1n0, L1n1,
                                     fn_lin1, sc_node, nacc);
  k_edge_pre<<<3750, 256, 0, stream>>>(f_edge, node_emb, length_emb, esrc, edst,
                                       W_sce0, W_sce1, L1e0, L1e1, fe_lin1, sc_edge);
  k_hidden<<<469, 128, 0, stream>>>(length_emb, Wm1, hbuf);
  size_t smemC = 34048 * sizeof(float);   // ~133 KB dynamic LDS
  k_edge_main<<<469, 128, smemC, stream>>>(sh, esrc, edst, fn_lin1, fe_lin1, sc_edge,
                                           hbuf, Wm2, L2e0, L2e1, nacc, out + 400000);
  k_node_post<<<40, 128, 0, stream>>>(nacc, sc_node, L2n0, L2n1, out);
}